// GraphSAGE_60859686584876
// MI455X (gfx1250) — compile-verified
//
#include <hip/hip_runtime.h>

typedef float v2f __attribute__((ext_vector_type(2)));
typedef float v8f __attribute__((ext_vector_type(8)));

#define N_NODES   50000
#define N_EDGES   800000
#define D_IN      128
#define HIDDEN    128
#define N_CLASSES 40

// ---------------------------------------------------------------------------
// Degree accumulation: one thread per edge, float atomic add on dst degree.
// ---------------------------------------------------------------------------
__global__ __launch_bounds__(256)
void sage_deg_kernel(const int* __restrict__ dst, float* __restrict__ deg, int E) {
    int i = blockIdx.x * blockDim.x + threadIdx.x;
    if (i < E) atomicAdd(&deg[dst[i]], 1.0f);
}

// deg -> 1/max(deg,1)   (so the GEMM can fuse the mean-divide into A loads)
__global__ __launch_bounds__(256)
void sage_invdeg_kernel(float* __restrict__ deg, int n) {
    int i = blockIdx.x * blockDim.x + threadIdx.x;
    if (i < n) deg[i] = 1.0f / fmaxf(deg[i], 1.0f);
}

// ---------------------------------------------------------------------------
// Scatter-sum of messages: thread = (edge, 4-feature chunk).
// x row is gathered as float4 (L2-resident), accumulated with float atomics.
// ---------------------------------------------------------------------------
__global__ __launch_bounds__(256)
void sage_scatter_kernel(const float* __restrict__ x,
                         const int* __restrict__ src,
                         const int* __restrict__ dst,
                         float* __restrict__ agg, int E) {
    int idx = blockIdx.x * blockDim.x + threadIdx.x;   // < E*32 = 25.6M
    if (idx >= E * 32) return;
    int e = idx >> 5;
    int g = (idx & 31) << 2;                            // feature offset 0..124
    int s = src[e];
    int d = dst[e];
    const float4 v = *(const float4*)(x + (size_t)s * D_IN + g);
    float* o = agg + (size_t)d * D_IN + g;
    atomicAdd(o + 0, v.x);
    atomicAdd(o + 1, v.y);
    atomicAdd(o + 2, v.z);
    atomicAdd(o + 3, v.w);
}

// ---------------------------------------------------------------------------
// Fused SAGE layer GEMM:
//   out[m,n] = relu?( (agg[m,:]*invdeg[m]) @ Wl + Aself[m,:] @ Wr + bias[n] )
// K fixed at 128. One wave computes a 16x16 output tile with
// V_WMMA_F32_16X16X4_F32, chaining both matrix products into one accumulator.
// Block = 4 waves sharing one 16-column slice of Wl/Wr staged in LDS.
// ---------------------------------------------------------------------------
__global__ __launch_bounds__(128)
void sage_gemm_kernel(const float* __restrict__ Aagg,
                      const float* __restrict__ invdeg,
                      const float* __restrict__ Aself,
                      const float* __restrict__ Wl,
                      const float* __restrict__ Wr,
                      const float* __restrict__ bias,
                      float* __restrict__ out,
                      int M, int Nout, int do_relu) {
    const int K = 128;
    __shared__ float Wl_s[K * 16];
    __shared__ float Wr_s[K * 16];

    const int col_base = blockIdx.y * 16;

    // Stage W slices [K x 16] into LDS (zero-pad columns beyond Nout).
    for (int i = threadIdx.x; i < K * 16; i += blockDim.x) {
        int k  = i >> 4;
        int c  = i & 15;
        int gc = col_base + c;
        float wl = 0.0f, wr = 0.0f;
        if (gc < Nout) {
            wl = Wl[(size_t)k * Nout + gc];
            wr = Wr[(size_t)k * Nout + gc];
        }
        Wl_s[i] = wl;
        Wr_s[i] = wr;
    }
    __syncthreads();

    const int wave = threadIdx.x >> 5;
    const int lane = threadIdx.x & 31;
    const int half = lane >> 4;     // 0: lanes 0-15, 1: lanes 16-31
    const int lidx = lane & 15;

    const int rowTiles = M >> 4;                 // 50000/16 = 3125, exact
    const int row_tile = blockIdx.x * 4 + wave;  // wave-uniform guard
    if (row_tile >= rowTiles) return;
    const int row_base = row_tile * 16;

    // A-matrix f32 16x4 layout: lane (half,lidx) supplies
    //   VGPR0 = A[row_base+lidx][kk + 2*half], VGPR1 = ... +1
    // B-matrix 4x16 mirrors it: VGPRv = W[kk + 2*half + v][col_base+lidx]
    const int    arow    = row_base + lidx;
    const float  scale   = invdeg[arow];
    const float* aggRow  = Aagg  + (size_t)arow * K;
    const float* selfRow = Aself + (size_t)arow * K;

    v8f acc = {0.f, 0.f, 0.f, 0.f, 0.f, 0.f, 0.f, 0.f};

    #pragma unroll 4
    for (int kk = 0; kk < K; kk += 4) {
        const int k0 = kk + 2 * half;

        v2f a_agg, a_self, b_l, b_r;
        a_agg.x  = aggRow[k0]     * scale;
        a_agg.y  = aggRow[k0 + 1] * scale;
        a_self.x = selfRow[k0];
        a_self.y = selfRow[k0 + 1];
        b_l.x = Wl_s[k0 * 16 + lidx];
        b_l.y = Wl_s[(k0 + 1) * 16 + lidx];
        b_r.x = Wr_s[k0 * 16 + lidx];
        b_r.y = Wr_s[(k0 + 1) * 16 + lidx];

        // acc = a_agg x b_l + acc ; acc = a_self x b_r + acc
        acc = __builtin_amdgcn_wmma_f32_16x16x4_f32(
                  false, a_agg, false, b_l, (short)0, acc, false, false);
        acc = __builtin_amdgcn_wmma_f32_16x16x4_f32(
                  false, a_self, false, b_r, (short)0, acc, false, false);
    }

    // C/D layout: VGPR v -> row = row_base + v + 8*half, col = col_base+lidx
    const int ocol = col_base + lidx;
    if (ocol < Nout) {
        const float bv = bias[ocol];
        #pragma unroll
        for (int v = 0; v < 8; ++v) {
            const int orow = row_base + v + 8 * half;
            float val = acc[v] + bv;
            if (do_relu) val = fmaxf(val, 0.0f);
            out[(size_t)orow * Nout + ocol] = val;
        }
    }
}

// ---------------------------------------------------------------------------
// Host-side orchestration (graph-capture safe: only *Async ops on `stream`).
// ---------------------------------------------------------------------------
extern "C" void kernel_launch(void* const* d_in, const int* in_sizes, int n_in,
                              void* d_out, int out_size, void* d_ws, size_t ws_size,
                              hipStream_t stream) {
    const float* x   = (const float*)d_in[0];
    const int*   ei  = (const int*)d_in[1];   // [2, E] int32
    const float* W1l = (const float*)d_in[2];
    const float* b1  = (const float*)d_in[3];
    const float* W1r = (const float*)d_in[4];
    const float* W2l = (const float*)d_in[5];
    const float* b2  = (const float*)d_in[6];
    const float* W2r = (const float*)d_in[7];
    float*       out = (float*)d_out;

    const int* src = ei;             // edge_index[0]
    const int* dst = ei + N_EDGES;   // edge_index[1]

    // Workspace layout (floats): agg1 | h | agg2 | invdeg  (~77 MB)
    float* agg1   = (float*)d_ws;
    float* h      = agg1 + (size_t)N_NODES * D_IN;
    float* agg2   = h    + (size_t)N_NODES * HIDDEN;
    float* invdeg = agg2 + (size_t)N_NODES * HIDDEN;

    hipMemsetAsync(agg1,   0, sizeof(float) * (size_t)N_NODES * D_IN,   stream);
    hipMemsetAsync(agg2,   0, sizeof(float) * (size_t)N_NODES * HIDDEN, stream);
    hipMemsetAsync(invdeg, 0, sizeof(float) * (size_t)N_NODES,          stream);

    // Degrees (shared by both layers), then invert for fused mean.
    sage_deg_kernel<<<(N_EDGES + 255) / 256, 256, 0, stream>>>(dst, invdeg, N_EDGES);
    sage_invdeg_kernel<<<(N_NODES + 255) / 256, 256, 0, stream>>>(invdeg, N_NODES);

    const int sc_blocks = (N_EDGES * 32 + 255) / 256;   // 100000 blocks

    // Layer 1: scatter-sum x -> agg1 ; h = relu(mean(agg1)@W1l + x@W1r + b1)
    sage_scatter_kernel<<<sc_blocks, 256, 0, stream>>>(x, src, dst, agg1, N_EDGES);
    {
        dim3 grid((N_NODES / 16 + 3) / 4, HIDDEN / 16);  // (782, 8)
        sage_gemm_kernel<<<grid, 128, 0, stream>>>(
            agg1, invdeg, x, W1l, W1r, b1, h, N_NODES, HIDDEN, 1);
    }

    // Layer 2: scatter-sum h -> agg2 ; out = mean(agg2)@W2l + h@W2r + b2
    sage_scatter_kernel<<<sc_blocks, 256, 0, stream>>>(h, src, dst, agg2, N_EDGES);
    {
        dim3 grid((N_NODES / 16 + 3) / 4, (N_CLASSES + 15) / 16);  // (782, 3)
        sage_gemm_kernel<<<grid, 128, 0, stream>>>(
            agg2, invdeg, h, W2l, W2r, b2, out, N_NODES, N_CLASSES, 0);
    }
}